// RGCN_orig_4037269258410
// MI455X (gfx1250) — compile-verified
//
#include <hip/hip_runtime.h>

typedef __attribute__((ext_vector_type(2))) float v2f;
typedef __attribute__((ext_vector_type(8))) float v8f;

namespace {
constexpr int kNU = 50000, kNN = 20000, kNS = 2000;
constexpr int kDH = 128, kDO = 64;
constexpr int kEF = 500000, kEP = 300000, kEPB = 300000, kEUS = 150000, kEUB = 150000;
constexpr float kSlope = 0.01f;
}

__global__ void zero_f32(float* __restrict__ p, long n) {
  long i = (long)blockIdx.x * blockDim.x + threadIdx.x;
  long stride = (long)gridDim.x * blockDim.x;
  for (; i < n; i += stride) p[i] = 0.0f;
}

__global__ void degree_kernel(const int* __restrict__ src, const int* __restrict__ dst,
                              float* __restrict__ dout, float* __restrict__ din, int ne) {
  int e = blockIdx.x * blockDim.x + threadIdx.x;
  if (e < ne) {
    atomicAdd(&dout[src[e]], 1.0f);
    atomicAdd(&din[dst[e]], 1.0f);
  }
}

__global__ void rsqrt_finalize(float* __restrict__ p, int n) {
  int i = blockIdx.x * blockDim.x + threadIdx.x;
  if (i < n) p[i] = rsqrtf(fmaxf(p[i], 1.0f));
}

// out[M x N] = (A[M x 128] * rowScale[M]) @ W[128 x N] (+ bias[N])
// 4 waves/block, 64 rows/block; V_WMMA_F32_16X16X4_F32.
// W staged in LDS in fragment-native pair layout:
//   for K-group g2 = (k/4)*2 + ((k>>1)&1), pair (W[k&~1][c], W[(k&~1)+1][c])
//   lives at dword g2*2N + ((2c) ^ ((g2&1)*32))  -> single aligned ds_load_b64
//   per B fragment; XOR-swizzle keeps lane halves on disjoint LDS bank sets.
template <int N>
__global__ __launch_bounds__(128) void gemm_wmma_f32(
    const float* __restrict__ A, const float* __restrict__ rowScale,
    const float* __restrict__ W, const float* __restrict__ bias,
    float* __restrict__ out, int M) {
  __shared__ float ldsW[128 * N];
  for (int i = threadIdx.x * 4; i < 128 * N; i += 128 * 4) {
    float4 wv = *(const float4*)&W[i];
    const int k = i / N;            // i%4==0 && N%4==0 -> all 4 share row k
    const int c = i % N;
    const int g2 = ((k >> 2) << 1) | ((k >> 1) & 1);
    const int lo = k & 1;
    const int sw = (g2 & 1) * 32;   // bank swizzle (flips dword bit 5)
    float* base = &ldsW[g2 * (2 * N)];
    base[(((c + 0) * 2 + lo) ^ sw)] = wv.x;
    base[(((c + 1) * 2 + lo) ^ sw)] = wv.y;
    base[(((c + 2) * 2 + lo) ^ sw)] = wv.z;
    base[(((c + 3) * 2 + lo) ^ sw)] = wv.w;
  }
  __syncthreads();

  const int wave = threadIdx.x >> 5;
  const int lane = threadIdx.x & 31;
  const int lrow = lane & 15;   // row-within-halfwave / column index
  const int hi   = lane >> 4;   // 0: K pair {k0,k0+1}, 1: {k0+2,k0+3}
  const int m0   = blockIdx.x * 64 + wave * 16;
  const int row  = m0 + lrow;
  const int rowc = row < M ? row : (M - 1);
  const float scale = rowScale ? rowScale[rowc] : 1.0f;
  const float* arow = A + (long)rowc * 128;

  v8f acc[N / 16] = {};

#pragma unroll 4
  for (int k0 = 0; k0 < 128; k0 += 4) {
    const int ka = k0 + hi * 2;
    float2 av = *(const float2*)&arow[ka];
    v2f a;
    a.x = av.x * scale;
    a.y = av.y * scale;
    const int g2 = ((k0 >> 2) << 1) | hi;
    const float* brow = &ldsW[g2 * (2 * N)];
    const int sw = hi * 32;
#pragma unroll
    for (int j = 0; j < N / 16; ++j) {
      v2f b = *(const v2f*)&brow[(((j * 16 + lrow) * 2) ^ sw)];
      acc[j] = __builtin_amdgcn_wmma_f32_16x16x4_f32(
          false, a, false, b, (short)0, acc[j], false, false);
    }
  }

#pragma unroll
  for (int j = 0; j < N / 16; ++j) {
    const int col = j * 16 + lrow;
    const float badd = bias ? bias[col] : 0.0f;
#pragma unroll
    for (int v = 0; v < 8; ++v) {
      const int r = m0 + v + hi * 8;
      if (r < M) out[(long)r * N + col] = acc[j][v] + badd;
    }
  }
}

// agg[dst[e]] += h[src[e]], 4 features per thread via float4 gather + f32 atomics.
__global__ void scatter_add_kernel(const float* __restrict__ h, const int* __restrict__ src,
                                   const int* __restrict__ dst, float* __restrict__ agg, int ne) {
  long idx = (long)blockIdx.x * blockDim.x + threadIdx.x;
  long total = (long)ne * 32;
  if (idx >= total) return;
  int e = (int)(idx >> 5);
  int q = (int)(idx & 31) << 2;
  int s = src[e], d = dst[e];
  float4 hv = *(const float4*)&h[(long)s * 128 + q];
  float* ap = &agg[(long)d * 128 + q];
  atomicAdd(ap + 0, hv.x);
  atomicAdd(ap + 1, hv.y);
  atomicAdd(ap + 2, hv.z);
  atomicAdd(ap + 3, hv.w);
}

__device__ __forceinline__ float lrelu(float v) { return v > 0.0f ? v : v * kSlope; }

// out = lrelu(0.5*((aggA*rsA + bA) + (aggB*rsB + bB)))
__global__ void combine2_kernel(float* __restrict__ out,
                                const float* __restrict__ aggA, const float* __restrict__ rsA,
                                const float* __restrict__ bA,
                                const float* __restrict__ aggB, const float* __restrict__ rsB,
                                const float* __restrict__ bB, int n) {
  long idx = (long)blockIdx.x * blockDim.x + threadIdx.x;
  long total = (long)n * 32;
  if (idx >= total) return;
  int node = (int)(idx >> 5);
  int q = (int)(idx & 31) << 2;
  float sa = rsA[node], sb = rsB[node];
  float4 a = *(const float4*)&aggA[(long)node * 128 + q];
  float4 b = *(const float4*)&aggB[(long)node * 128 + q];
  float4 ba = *(const float4*)&bA[q];
  float4 bb = *(const float4*)&bB[q];
  float4 r;
  r.x = lrelu(0.5f * (a.x * sa + ba.x + b.x * sb + bb.x));
  r.y = lrelu(0.5f * (a.y * sa + ba.y + b.y * sb + bb.y));
  r.z = lrelu(0.5f * (a.z * sa + ba.z + b.z * sb + bb.z));
  r.w = lrelu(0.5f * (a.w * sa + ba.w + b.w * sb + bb.w));
  *(float4*)&out[(long)node * 128 + q] = r;
}

// out = lrelu(aggA*rsA + bA)
__global__ void combine1_kernel(float* __restrict__ out,
                                const float* __restrict__ aggA, const float* __restrict__ rsA,
                                const float* __restrict__ bA, int n) {
  long idx = (long)blockIdx.x * blockDim.x + threadIdx.x;
  long total = (long)n * 32;
  if (idx >= total) return;
  int node = (int)(idx >> 5);
  int q = (int)(idx & 31) << 2;
  float sa = rsA[node];
  float4 a = *(const float4*)&aggA[(long)node * 128 + q];
  float4 ba = *(const float4*)&bA[q];
  float4 r;
  r.x = lrelu(a.x * sa + ba.x);
  r.y = lrelu(a.y * sa + ba.y);
  r.z = lrelu(a.z * sa + ba.z);
  r.w = lrelu(a.w * sa + ba.w);
  *(float4*)&out[(long)node * 128 + q] = r;
}

static inline int cdiv_i(long a, long b) { return (int)((a + b - 1) / b); }

extern "C" void kernel_launch(void* const* d_in, const int* in_sizes, int n_in,
                              void* d_out, int out_size, void* d_ws, size_t ws_size,
                              hipStream_t stream) {
  (void)in_sizes; (void)n_in; (void)out_size; (void)ws_size;

  const float* xu = (const float*)d_in[0];
  const float* xn = (const float*)d_in[1];
  const float* xs = (const float*)d_in[2];
  const float* W1 = (const float*)d_in[3];
  const float* b1 = (const float*)d_in[4];
  const float* W2 = (const float*)d_in[5];
  const float* b2 = (const float*)d_in[6];
  const float* Wu = (const float*)d_in[7];
  const float* bu = (const float*)d_in[8];
  const float* Wn = (const float*)d_in[9];
  const float* bn = (const float*)d_in[10];
  const float* Wsrc = (const float*)d_in[11];
  const float* bsrc = (const float*)d_in[12];
  const int* fs  = (const int*)d_in[13];
  const int* fd  = (const int*)d_in[14];
  const int* ps  = (const int*)d_in[15];
  const int* pd  = (const int*)d_in[16];
  const int* pbs = (const int*)d_in[17];
  const int* pbd = (const int*)d_in[18];
  const int* us  = (const int*)d_in[19];
  const int* ud  = (const int*)d_in[20];
  const int* ubs = (const int*)d_in[21];
  const int* ubd = (const int*)d_in[22];

  float* w = (float*)d_ws;
  size_t off = 0;
  auto take = [&](size_t n) { float* p = w + off; off += n; return p; };
  // degree / rsqrt arrays (contiguous for single zero + finalize pass)
  float* rs_dout_f  = take(kNU);
  float* rs_din_f   = take(kNU);
  float* rs_dout_p  = take(kNU);
  float* rs_din_p   = take(kNN);
  float* rs_dout_pb = take(kNN);
  float* rs_din_pb  = take(kNU);
  float* rs_dout_us = take(kNS);
  float* rs_din_ud  = take(kNN);
  float* rs_dout_ub = take(kNN);
  float* rs_din_ub  = take(kNS);
  const size_t degTotal = off;
  // per-relation aggregation buffers (contiguous for single zero pass)
  float* aggF  = take((size_t)kNU * 128);
  float* aggPB = take((size_t)kNU * 128);
  float* aggP  = take((size_t)kNN * 128);
  float* aggUD = take((size_t)kNN * 128);
  float* aggUB = take((size_t)kNS * 128);
  const size_t aggTotal = off - degTotal;
  float* hbuf = take((size_t)kNU * 128);   // transform scratch (max rows)
  float* h1u  = take((size_t)kNU * 128);
  float* h1n  = take((size_t)kNN * 128);
  float* h1s  = take((size_t)kNS * 128);

  float* ou  = (float*)d_out;
  float* on  = ou + (size_t)kNU * kDO;
  float* osrc = on + (size_t)kNN * kDO;
  float* hu2 = osrc + (size_t)kNS * kDO;
  float* hn2 = hu2 + (size_t)kNU * kDH;
  float* hs2 = hn2 + (size_t)kNN * kDH;

  // ---- degrees (same graph for both layers) ----
  zero_f32<<<1024, 256, 0, stream>>>(rs_dout_f, (long)degTotal);
  degree_kernel<<<cdiv_i(kEF, 256), 256, 0, stream>>>(fs, fd, rs_dout_f, rs_din_f, kEF);
  degree_kernel<<<cdiv_i(kEP, 256), 256, 0, stream>>>(ps, pd, rs_dout_p, rs_din_p, kEP);
  degree_kernel<<<cdiv_i(kEPB, 256), 256, 0, stream>>>(pbs, pbd, rs_dout_pb, rs_din_pb, kEPB);
  degree_kernel<<<cdiv_i(kEUS, 256), 256, 0, stream>>>(us, ud, rs_dout_us, rs_din_ud, kEUS);
  degree_kernel<<<cdiv_i(kEUB, 256), 256, 0, stream>>>(ubs, ubd, rs_dout_ub, rs_din_ub, kEUB);
  rsqrt_finalize<<<cdiv_i((long)degTotal, 256), 256, 0, stream>>>(rs_dout_f, (int)degTotal);

  auto gconv = [&](const float* X, int nsrc, const float* rsOut, const float* Wp,
                   const int* esrc, const int* edst, int ne, float* agg) {
    gemm_wmma_f32<128><<<cdiv_i(nsrc, 64), 128, 0, stream>>>(X, rsOut, Wp, nullptr, hbuf, nsrc);
    scatter_add_kernel<<<cdiv_i((long)ne * 32, 256), 256, 0, stream>>>(hbuf, esrc, edst, agg, ne);
  };

  auto layer = [&](const float* iu, const float* inw, const float* is,
                   const float* Wl, const float* bl,
                   float* houtU, float* houtN, float* houtS) {
    zero_f32<<<4096, 256, 0, stream>>>(aggF, (long)aggTotal);
    gconv(iu,  kNU, rs_dout_f,  Wl + 0 * 128 * 128, fs,  fd,  kEF,  aggF);
    gconv(iu,  kNU, rs_dout_p,  Wl + 1 * 128 * 128, ps,  pd,  kEP,  aggP);
    gconv(inw, kNN, rs_dout_pb, Wl + 2 * 128 * 128, pbs, pbd, kEPB, aggPB);
    gconv(is,  kNS, rs_dout_us, Wl + 3 * 128 * 128, us,  ud,  kEUS, aggUD);
    gconv(inw, kNN, rs_dout_ub, Wl + 4 * 128 * 128, ubs, ubd, kEUB, aggUB);
    combine2_kernel<<<cdiv_i((long)kNU * 32, 256), 256, 0, stream>>>(
        houtU, aggF, rs_din_f, bl + 0 * 128, aggPB, rs_din_pb, bl + 2 * 128, kNU);
    combine2_kernel<<<cdiv_i((long)kNN * 32, 256), 256, 0, stream>>>(
        houtN, aggP, rs_din_p, bl + 1 * 128, aggUD, rs_din_ud, bl + 3 * 128, kNN);
    combine1_kernel<<<cdiv_i((long)kNS * 32, 256), 256, 0, stream>>>(
        houtS, aggUB, rs_din_ub, bl + 4 * 128, kNS);
  };

  layer(xu, xn, xs, W1, b1, h1u, h1n, h1s);
  layer(h1u, h1n, h1s, W2, b2, hu2, hn2, hs2);

  gemm_wmma_f32<64><<<cdiv_i(kNU, 64), 128, 0, stream>>>(hu2, nullptr, Wu, bu, ou, kNU);
  gemm_wmma_f32<64><<<cdiv_i(kNN, 64), 128, 0, stream>>>(hn2, nullptr, Wn, bn, on, kNN);
  gemm_wmma_f32<64><<<cdiv_i(kNS, 64), 128, 0, stream>>>(hs2, nullptr, Wsrc, bsrc, osrc, kNS);
}